// InstanceSegmentationLoss_72662256713776
// MI455X (gfx1250) — compile-verified
//
#include <hip/hip_runtime.h>
#include <hip/hip_bf16.h>

typedef __attribute__((ext_vector_type(2))) float v2f;
typedef __attribute__((ext_vector_type(8))) float v8f;

#define K_INST 64
#define LAMBDA_SEP 300.0f
#define MASK_BIG 1.0e30f   // rows biased by this contribute ~3e-28 per pixel: negligible

// Workspace float layout:
//   stats : per image stride 320  -> [cnt(64) | sx(64) | sy(64) | sz(64) | hs(64)]
//   bmat  : base B*320, per image stride 256 -> rows {-2mx, -2my, -2mz, |m|^2}[64]
//   sep   : base B*576, per image stride 128 -> [S1(64) | S2 | pad]

__device__ __forceinline__ float huber1(float x, float c) {
    float d = fabsf(x - c);
    return (d < 1.0f) ? 0.5f * d * d : d - 0.5f;
}

__global__ void k_init(float* ws, int nfloats, float* out) {
    int i = blockIdx.x * blockDim.x + threadIdx.x;
    if (i < nfloats) ws[i] = 0.0f;
    if (i == 0) out[0] = 0.0f;
}

__global__ void k_stats(const float* __restrict__ pred,
                        const float* __restrict__ targ,
                        float* __restrict__ ws, int N) {
    __shared__ float s_cnt[K_INST], s_sx[K_INST], s_sy[K_INST], s_sz[K_INST], s_hs[K_INST];
    int t = threadIdx.x;
    for (int k = t; k < K_INST; k += blockDim.x) {
        s_cnt[k] = 0.0f; s_sx[k] = 0.0f; s_sy[k] = 0.0f; s_sz[k] = 0.0f; s_hs[k] = 0.0f;
    }
    __syncthreads();
    int b = blockIdx.y;
    const float* P = pred + (size_t)b * 3 * N;
    const float* T = targ + (size_t)b * 3 * N;
    int stride = gridDim.x * blockDim.x;
    for (int i = blockIdx.x * blockDim.x + t; i < N; i += stride) {
        float x = P[i], y = P[i + N], z = P[i + 2 * N];
        int lbl = __float2int_rn(T[i] * 0.25f);
        lbl = lbl < 0 ? 0 : (lbl > 63 ? 63 : lbl);
        float c = (lbl == 0) ? 0.0f : 255.0f;
        float h = huber1(x, c) + huber1(y, c) + huber1(z, c);
        atomicAdd(&s_cnt[lbl], 1.0f);
        atomicAdd(&s_sx[lbl], x);
        atomicAdd(&s_sy[lbl], y);
        atomicAdd(&s_sz[lbl], z);
        atomicAdd(&s_hs[lbl], h);
    }
    __syncthreads();
    float* S = ws + (size_t)b * 320;
    for (int k = t; k < K_INST; k += blockDim.x) {
        atomicAdd(&S[k],        s_cnt[k]);
        atomicAdd(&S[64 + k],   s_sx[k]);
        atomicAdd(&S[128 + k],  s_sy[k]);
        atomicAdd(&S[192 + k],  s_sz[k]);
        atomicAdd(&S[256 + k],  s_hs[k]);
    }
}

__global__ void k_means(float* ws, int B) {
    int t = blockIdx.x * blockDim.x + threadIdx.x;
    if (t >= B * K_INST) return;
    int b = t / K_INST, k = t % K_INST;
    const float* S = ws + (size_t)b * 320;
    float cnt  = S[k];
    float safe = fmaxf(cnt, 1.0f);
    float mx = S[64 + k] / safe, my = S[128 + k] / safe, mz = S[192 + k] / safe;
    float* Bm = ws + (size_t)B * 320 + (size_t)b * 256;
    Bm[k]        = -2.0f * mx;
    Bm[64 + k]   = -2.0f * my;
    Bm[128 + k]  = -2.0f * mz;
    Bm[192 + k]  = mx * mx + my * my + mz * mz;
}

// Separation term via V_WMMA_F32_16X16X4_F32:
//   1 + D[m][k] = (1 + |p_m|^2) + (p,1) . (-2m_k, |m_k|^2)
// Background masking is folded into the C-fragment: rows NOT in the wanted set
// get C += 1e30, so after rcp they contribute ~0 and the per-element epilogue
// is exactly v_rcp_f32 + v_add_f32.  A 5th WMMA with the opposite bias handles
// the instance-0 vs non-background term.
__global__ void __launch_bounds__(256) k_sep(const float* __restrict__ pred,
                                             const float* __restrict__ targ,
                                             float* __restrict__ ws, int N, int B) {
    int b      = blockIdx.y;
    int lane   = threadIdx.x & 31;
    int wave   = threadIdx.x >> 5;
    int wid    = blockIdx.x * (blockDim.x >> 5) + wave;
    int nwaves = gridDim.x * (blockDim.x >> 5);
    int ntiles = N >> 4;                       // N is a multiple of 16 (768*768)
    const float* P  = pred + (size_t)b * 3 * N;
    const float* T  = targ + (size_t)b * 3 * N;
    const float* Bm = ws + (size_t)B * 320 + (size_t)b * 256;

    int m     = lane & 15;                     // pixel row / instance column within tile
    int rbase = (lane & 16) >> 1;              // rows 0..7 (lanes 0-15) or 8..15 (lanes 16-31)
    int krow  = (lane < 16) ? 0 : 2;           // B-frag K rows per lane half

    v2f bfrag[4];
#pragma unroll
    for (int g = 0; g < 4; ++g) {
        int col = 16 * g + m;
        bfrag[g][0] = Bm[krow * 64 + col];
        bfrag[g][1] = Bm[(krow + 1) * 64 + col];
    }

    v8f acc[4] = {};
    float s2 = 0.0f;

    for (int tile = wid; tile < ntiles; tile += nwaves) {
        int px = tile * 16 + m;
        int nxt = px + nwaves * 16;
        if (nxt < N) {                          // warm caches for the next tile
            __builtin_prefetch(&P[nxt], 0, 1);
            __builtin_prefetch(&T[nxt], 0, 1);
        }
        float x = P[px], y = P[px + N], z = P[px + 2 * N];
        bool bg = __float2int_rn(T[px] * 0.25f) == 0;
        float q  = 1.0f + x * x + y * y + z * z;       // 1 + |p|^2
        float qm = bg ? q : q + MASK_BIG;              // keep background rows
        float qn = bg ? q + MASK_BIG : q;              // keep non-background rows

        v2f a;                                  // 16x4 f32 A layout (wave32)
        a[0] = (lane < 16) ? x : z;
        a[1] = (lane < 16) ? y : 1.0f;

        v8f cm, cn;                             // transposed row biases
#pragma unroll
        for (int e = 0; e < 8; ++e) {
            cm[e] = __shfl(qm, e + rbase, 32);
            cn[e] = __shfl(qn, e + rbase, 32);
        }

#pragma unroll
        for (int g = 0; g < 4; ++g) {
            v8f d = __builtin_amdgcn_wmma_f32_16x16x4_f32(
                false, a, false, bfrag[g], (short)0, cm, false, false);
#pragma unroll
            for (int e = 0; e < 8; ++e)
                acc[g][e] += __builtin_amdgcn_rcpf(d[e]);   // Σ_bg 1/(1+D)
        }

        // instance 0 vs non-background pixels (only column 0 consumed)
        v8f d5 = __builtin_amdgcn_wmma_f32_16x16x4_f32(
            false, a, false, bfrag[0], (short)0, cn, false, false);
        if (m == 0) {
#pragma unroll
            for (int e = 0; e < 8; ++e)
                s2 += __builtin_amdgcn_rcpf(d5[e]);
        }
    }

    float* S = ws + (size_t)B * (320 + 256) + (size_t)b * 128;
#pragma unroll
    for (int g = 0; g < 4; ++g) {
        float s = 0.0f;
#pragma unroll
        for (int e = 0; e < 8; ++e) s += acc[g][e];
        atomicAdd(&S[16 * g + m], LAMBDA_SEP * s);   // lanes c and c+16 fold into column c
    }
    if (m == 0) atomicAdd(&S[64], LAMBDA_SEP * s2);
}

__global__ void k_final(const float* __restrict__ ws,
                        const unsigned char* __restrict__ no_bg,
                        float* __restrict__ out, int N, int B) {
    __shared__ float sterm[256];
    __shared__ float sincl[256];
    int t = threadIdx.x;
    float term = 0.0f, incl = 0.0f;
    if (t < B * K_INST) {
        int b = t >> 6, k = t & 63;
        const float* S   = ws + (size_t)b * 320;
        const float* SEP = ws + (size_t)B * (320 + 256) + (size_t)b * 128;
        float cnt   = S[k];
        float safe  = fmaxf(cnt, 1.0f);
        float var   = S[256 + k] / (3.0f * safe);
        float bgcnt = S[0];
        float nbg   = (float)N - bgcnt;
        float sep   = (k == 0) ? SEP[64] / fmaxf(nbg, 1.0f)
                               : SEP[k] / fmaxf(bgcnt, 1.0f);
        float w = 10.0f * rsqrtf(safe);
        bool present = cnt > 0.0f;
        bool inc = (k == 0) ? (present && (no_bg[b] == 0)) : present;
        incl = inc ? 1.0f : 0.0f;
        term = incl * (var + w * sep);
    }
    sterm[t] = term;
    sincl[t] = incl;
    __syncthreads();
    for (int off = 32; off >= 1; off >>= 1) {
        if ((t & 63) < off) {
            sterm[t] += sterm[t + off];
            sincl[t] += sincl[t + off];
        }
        __syncthreads();
    }
    if (t == 0) {
        float L = 0.0f;
        for (int b = 0; b < B; ++b)
            L += sterm[b * 64] / fmaxf(sincl[b * 64], 1.0f);
        out[0] = L / (float)B;
    }
}

extern "C" void kernel_launch(void* const* d_in, const int* in_sizes, int n_in,
                              void* d_out, int out_size, void* d_ws, size_t ws_size,
                              hipStream_t stream) {
    const float* pred = (const float*)d_in[0];
    const float* targ = (const float*)d_in[1];
    const unsigned char* no_bg = (const unsigned char*)d_in[2];
    float* out = (float*)d_out;
    float* ws  = (float*)d_ws;

    int B = in_sizes[2];                  // batch = #no_bg flags
    int N = in_sizes[0] / (3 * B);        // pixels per image

    int nfloats = B * (320 + 256 + 128);
    k_init<<<(nfloats + 255) / 256, 256, 0, stream>>>(ws, nfloats, out);
    k_stats<<<dim3(192, B), 256, 0, stream>>>(pred, targ, ws, N);
    k_means<<<(B * K_INST + 255) / 256, 256, 0, stream>>>(ws, B);
    k_sep<<<dim3(96, B), 256, 0, stream>>>(pred, targ, ws, N, B);
    k_final<<<1, 256, 0, stream>>>(ws, no_bg, out, N, B);
}